// LbpLayer_50096498540818
// MI455X (gfx1250) — compile-verified
//
#include <hip/hip_runtime.h>
#include <cstdint>

#ifndef __has_builtin
#define __has_builtin(x) 0
#endif

#if __has_builtin(__builtin_amdgcn_global_load_async_to_lds_b128)
#define LBP_ASYNC_LDS 1
#else
#define LBP_ASYNC_LDS 0
#endif

typedef int v4i __attribute__((ext_vector_type(4)));
typedef __attribute__((address_space(1))) v4i GInt4;  // global int4
typedef __attribute__((address_space(3))) v4i LInt4;  // LDS int4

namespace {
constexpr int Bn = 8, Cn = 64, Hn = 224, Wn = 224;
constexpr int TH    = 8;            // output rows per block
constexpr int RPT   = 2;            // output rows per thread
constexpr int BY    = TH / RPT;     // blockDim.y = 4 -> 896 threads = 28 wave32
constexpr int LROWS = TH + 2;       // 10 rows staged in LDS (incl. halo)
constexpr int QROW  = Wn / 4;       // 56 float4 chunks per row
constexpr int NTASK = LROWS * QROW; // 560 float4 async copies per block
}

__device__ __forceinline__ float sgn_mul(float c, float n, float e) {
  float d = c - n;
  float s = (float)(d > 0.0f) - (float)(d < 0.0f); // sign(d) in {-1,0,1}
  return s * e;
}

__global__ __launch_bounds__(Wn * BY)
void lbp_kernel(const float* __restrict__ x, const float* __restrict__ w,
                float* __restrict__ out) {
  __shared__ float tile[LROWS * Wn]; // 10 rows * 224 floats = 8960 B
  __shared__ float ew[8];

  const int tx    = threadIdx.x;         // 0..223 (column)
  const int ty    = threadIdx.y;         // 0..3
  const int tid   = ty * Wn + tx;        // 0..895
  const int h0    = blockIdx.x * TH;     // first output row of this tile
  const int plane = blockIdx.y;          // b*C + c
  const int ch    = plane & (Cn - 1);
  const float* xp = x + (size_t)plane * (size_t)(Hn * Wn);
  float* op       = out + (size_t)plane * (size_t)(Hn * Wn);

  // Per-channel weights: exp(w[ch, k]), k = 0..7, once per block.
  if (tid < 8) ew[tid] = expf(w[ch * 8 + tid]);

  // Async-stage rows (h0-1 .. h0+TH) of this plane into LDS (ASYNCcnt path).
  if (tid < NTASK) {
    const int jr = tid / QROW;           // LDS row 0..9
    const int q  = tid % QROW;           // float4 index within row
    const int hg = h0 - 1 + jr;          // global row
    if (hg >= 0 && hg < Hn) {
#if LBP_ASYNC_LDS
      __builtin_amdgcn_global_load_async_to_lds_b128(
          (GInt4*)(xp + hg * Wn + q * 4),
          (LInt4*)(&tile[jr * Wn + q * 4]),
          /*offset=*/0, /*cpol=*/0);
#else
      const float4 v = *(const float4*)(xp + hg * Wn + q * 4);
      *(float4*)(&tile[jr * Wn + q * 4]) = v;
#endif
    }
  }
#if LBP_ASYNC_LDS
#if __has_builtin(__builtin_amdgcn_s_wait_asynccnt)
  __builtin_amdgcn_s_wait_asynccnt(0);
#else
  asm volatile("s_wait_asynccnt 0" ::: "memory");
#endif
#endif
  __syncthreads();

  const float e0 = ew[0], e1 = ew[1], e2 = ew[2], e3 = ew[3];
  const float e4 = ew[4], e5 = ew[5], e6 = ew[6], e7 = ew[7];

#pragma unroll
  for (int rep = 0; rep < RPT; ++rep) {
    const int lr = ty + rep * BY;        // local output row 0..7
    const int h  = h0 + lr;              // global output row
    const float* r0 = &tile[lr * Wn];    // global row h-1 (LDS row lr)
    const float* r1 = r0 + Wn;           // global row h
    const float* r2 = r1 + Wn;           // global row h+1
    const float  c  = r1[tx];

    float o = c;                          // border: y zero-padded -> out = x
    if (h > 0 && h < Hn - 1 && tx > 0 && tx < Wn - 1) {
      // neighbor order matches reference: [TL, T, TR, L, BL, B, BR, R]
      float s = sgn_mul(c, r0[tx - 1], e0);
      s += sgn_mul(c, r0[tx    ], e1);
      s += sgn_mul(c, r0[tx + 1], e2);
      s += sgn_mul(c, r1[tx - 1], e3);
      s += sgn_mul(c, r2[tx - 1], e4);
      s += sgn_mul(c, r2[tx    ], e5);
      s += sgn_mul(c, r2[tx + 1], e6);
      s += sgn_mul(c, r1[tx + 1], e7);
      o = c + s;
    }
    op[(size_t)h * Wn + tx] = o;          // coalesced b32 row store
  }
}

extern "C" void kernel_launch(void* const* d_in, const int* in_sizes, int n_in,
                              void* d_out, int out_size, void* d_ws, size_t ws_size,
                              hipStream_t stream) {
  (void)in_sizes; (void)n_in; (void)d_ws; (void)ws_size; (void)out_size;
  const float* x  = (const float*)d_in[0];
  const float* w  = (const float*)d_in[1];
  float*      out = (float*)d_out;
  dim3 grid(Hn / TH, Bn * Cn, 1);   // (28, 512)
  dim3 block(Wn, BY, 1);            // (224, 4) = 896 threads = 28 wave32
  hipLaunchKernelGGL(lbp_kernel, grid, block, 0, stream, x, w, out);
}